// TGATConv_41008347742502
// MI455X (gfx1250) — compile-verified
//
#include <hip/hip_runtime.h>
#include <hip/hip_bf16.h>

typedef __attribute__((ext_vector_type(2))) float v2f;
typedef __attribute__((ext_vector_type(8))) float v8f;

#define NEG_SLOPE 0.2f

// ---------- monotone float <-> uint encoding for atomicMax over signed floats ----------
__device__ __forceinline__ unsigned fenc(float f) {
  int i = __float_as_int(f);
  return (i >= 0) ? ((unsigned)i | 0x80000000u) : ~(unsigned)i;
}
__device__ __forceinline__ float fdec(unsigned u) {
  int i = (u & 0x80000000u) ? (int)(u & 0x7fffffffu) : ~(int)u;
  return __int_as_float(i);
}

// ---------------------------------------------------------------------------
// Kernel 0: fold attn through W_fc and W_type:
//   vl[j,h] = sum_f W_fc[h*32+f, j] * attn_l[h,f]   (vr likewise)
//   plr[c, n] = sum_j W_type[j, c] * vlr[j, n]      n: 0..3 = el heads, 4..7 = er heads
// and store plr pre-swizzled into WMMA-B fragment order:
//   fragB[t*64 + lane*2 + v] = plr[4t + (lane>=16?2:0) + v][lane%16]  (0 for n>=8)
// ---------------------------------------------------------------------------
__global__ void precompute_kernel(const float* __restrict__ W_type,  // [64,100]
                                  const float* __restrict__ W_fc,    // [128,64]
                                  const float* __restrict__ attn_l,  // [128]
                                  const float* __restrict__ attn_r,  // [128]
                                  float* __restrict__ fragB) {       // [25*64]
  __shared__ float vlr[64][8];
  int tid = threadIdx.x;  // 256 threads
  for (int idx = tid; idx < 512; idx += 256) {
    int j = idx >> 3, c = idx & 7;
    const float* attn = (c < 4) ? attn_l : attn_r;
    int h = c & 3;
    float s = 0.f;
    #pragma unroll 8
    for (int f = 0; f < 32; ++f)
      s += W_fc[(h * 32 + f) * 64 + j] * attn[h * 32 + f];
    vlr[j][c] = s;
  }
  __syncthreads();
  for (int idx = tid; idx < 1600; idx += 256) {
    int t = idx >> 6;           // K-step
    int r = idx & 63;           // lane*2 + v
    int lane = r >> 1, v = r & 1;
    int n = lane & 15;
    int k = 4 * t + ((lane >> 4) << 1) + v;   // 0..99
    float s = 0.f;
    if (n < 8) {
      #pragma unroll 8
      for (int j = 0; j < 64; ++j)
        s += W_type[j * 100 + k] * vlr[j][n];
    }
    fragB[idx] = s;
  }
}

// ---------------------------------------------------------------------------
// Kernel 1: zero the accumulation buffers
// ---------------------------------------------------------------------------
__global__ void init_kernel(float* __restrict__ agg, float* __restrict__ denom,
                            unsigned* __restrict__ maxu, long nAgg, int nND) {
  long i = (long)blockIdx.x * 256 + threadIdx.x;
  long stride = (long)gridDim.x * 256;
  for (long k = i; k < nAgg; k += stride) agg[k] = 0.f;
  for (long k = i; k < nND; k += stride) { denom[k] = 0.f; maxu[k] = 0u; }
}

// ---------------------------------------------------------------------------
// Kernel 2: el/er = feat[N,100] @ plr[100,8] via V_WMMA_F32_16X16X4_F32.
// One wave per 16-node tile. EXEC all-ones at every WMMA (per-wave early exit).
// ---------------------------------------------------------------------------
__global__ void node_elr_kernel(const float* __restrict__ feat,   // [N,100]
                                const float* __restrict__ fragB,  // [25*64]
                                float* __restrict__ el,           // [N,4]
                                float* __restrict__ er,           // [N,4]
                                int N) {
  int wave = (int)((blockIdx.x * blockDim.x + threadIdx.x) >> 5);
  int lane = threadIdx.x & 31;
  int ntiles = (N + 15) >> 4;
  if (wave >= ntiles) return;

  int m16 = lane & 15;
  int khalf = (lane >> 4) << 1;                 // lanes 16-31 hold K=2,3
  int row = (wave << 4) + m16;
  int rowc = row < N ? row : N - 1;
  const float* arow = feat + (long)rowc * 100 + khalf;
  bool rowok = row < N;

  v8f acc = {};
  #pragma unroll
  for (int t = 0; t < 25; ++t) {
    v2f a = *(const v2f*)(arow + 4 * t);        // 8B-aligned: (row*100 + even)*4
    if (!rowok) { a.x = 0.f; a.y = 0.f; }
    v2f b = *(const v2f*)(fragB + t * 64 + lane * 2);
    acc = __builtin_amdgcn_wmma_f32_16x16x4_f32(false, a, false, b, (short)0, acc,
                                                false, false);
  }
  // D(M,N): M = v + 8*(lane/16), N = lane%16.  N<4 -> el head N ; 4..7 -> er head N-4
  int n = lane & 15;
  int mhi = (lane >> 4) << 3;
  if (n < 8) {
    float* dstbuf = (n < 4) ? el : er;
    int hh = n & 3;
    #pragma unroll
    for (int v = 0; v < 8; ++v) {
      int m = (wave << 4) + mhi + v;
      if (m < N) dstbuf[m * 4 + hh] = acc[v];
    }
  }
}

// ---------------------------------------------------------------------------
// Kernel 3: per-edge logits e = leakyrelu(el[src]+er[dst]); atomic segment max
// ---------------------------------------------------------------------------
__global__ void edge_logits_kernel(const int* __restrict__ src, const int* __restrict__ dst,
                                   const float* __restrict__ el, const float* __restrict__ er,
                                   float* __restrict__ e, unsigned* __restrict__ maxu, int E) {
  int i = blockIdx.x * 256 + threadIdx.x;
  if (i >= E) return;
  int s = src[i], d = dst[i];
  float4 l = *(const float4*)(el + (long)s * 4);
  float4 r = *(const float4*)(er + (long)d * 4);
  float4 v;
  v.x = l.x + r.x; v.x = v.x >= 0.f ? v.x : NEG_SLOPE * v.x;
  v.y = l.y + r.y; v.y = v.y >= 0.f ? v.y : NEG_SLOPE * v.y;
  v.z = l.z + r.z; v.z = v.z >= 0.f ? v.z : NEG_SLOPE * v.z;
  v.w = l.w + r.w; v.w = v.w >= 0.f ? v.w : NEG_SLOPE * v.w;
  *(float4*)(e + (long)i * 4) = v;
  unsigned* mb = maxu + (long)d * 4;
  atomicMax(mb + 0, fenc(v.x));
  atomicMax(mb + 1, fenc(v.y));
  atomicMax(mb + 2, fenc(v.z));
  atomicMax(mb + 3, fenc(v.w));
}

// ---------------------------------------------------------------------------
// Kernel 4: ex = exp(e - max[dst]); atomic segment sum into denom
// ---------------------------------------------------------------------------
__global__ void edge_exp_kernel(const int* __restrict__ dst, float* __restrict__ e,
                                const unsigned* __restrict__ maxu,
                                float* __restrict__ denom, int E) {
  int i = blockIdx.x * 256 + threadIdx.x;
  if (i >= E) return;
  int d = dst[i];
  float4 ev = *(const float4*)(e + (long)i * 4);
  uint4 mu = *(const uint4*)(maxu + (long)d * 4);
  float4 x;
  x.x = __expf(ev.x - fdec(mu.x));
  x.y = __expf(ev.y - fdec(mu.y));
  x.z = __expf(ev.z - fdec(mu.z));
  x.w = __expf(ev.w - fdec(mu.w));
  *(float4*)(e + (long)i * 4) = x;
  float* dn = denom + (long)d * 4;
  atomicAdd(dn + 0, x.x);
  atomicAdd(dn + 1, x.y);
  atomicAdd(dn + 2, x.z);
  atomicAdd(dn + 3, x.w);
}

// ---------------------------------------------------------------------------
// Kernel 5 (dominant): agg[dst,h,:] += node_feature[src,:] * a[h]
// Thread = (edge, dim); all 32 lanes of a wave share one edge (uniform scalars).
// ---------------------------------------------------------------------------
__global__ void edge_agg_kernel(const int* __restrict__ src, const int* __restrict__ dst,
                                const float* __restrict__ ex, const float* __restrict__ denom,
                                const float* __restrict__ nf,   // [N,64]
                                float* __restrict__ agg,        // [N,4,64]
                                long total) {
  long idx = (long)blockIdx.x * 256 + threadIdx.x;
  if (idx >= total) return;
  int dth = (int)(idx & 63);
  long eid = idx >> 6;
  int s = src[eid], d = dst[eid];
  float x = nf[(long)s * 64 + dth];
  float4 a4 = *(const float4*)(ex + eid * 4);
  float4 dn = *(const float4*)(denom + (long)d * 4);
  float* base = agg + (long)d * 256 + dth;
  atomicAdd(base + 0,   x * (a4.x / dn.x));
  atomicAdd(base + 64,  x * (a4.y / dn.y));
  atomicAdd(base + 128, x * (a4.z / dn.z));
  atomicAdd(base + 192, x * (a4.w / dn.w));
}

// ---------------------------------------------------------------------------
// Kernel 6: rst = agg[N*4,64] @ W_out.T[64,32] + bias, via WMMA f32 16x16x4.
// One wave per 16-row tile, two 16-col output tiles.
// ---------------------------------------------------------------------------
__global__ void out_gemm_kernel(const float* __restrict__ agg,    // [R,64]
                                const float* __restrict__ W_out,  // [32,64]
                                const float* __restrict__ bias,   // [128]
                                float* __restrict__ out,          // [R,32]
                                int R) {
  int wave = (int)((blockIdx.x * blockDim.x + threadIdx.x) >> 5);
  int lane = threadIdx.x & 31;
  int ntiles = (R + 15) >> 4;
  if (wave >= ntiles) return;

  int m16 = lane & 15;
  int khalf = (lane >> 4) << 1;
  int row = (wave << 4) + m16;
  int rowc = row < R ? row : R - 1;
  bool rowok = row < R;
  const float* arow = agg + (long)rowc * 64 + khalf;
  const float* b0row = W_out + m16 * 64 + khalf;          // cols f = 0..15
  const float* b1row = W_out + (16 + m16) * 64 + khalf;   // cols f = 16..31

  v8f acc0 = {}, acc1 = {};
  #pragma unroll
  for (int t = 0; t < 16; ++t) {
    v2f a = *(const v2f*)(arow + 4 * t);
    if (!rowok) { a.x = 0.f; a.y = 0.f; }
    v2f b0 = *(const v2f*)(b0row + 4 * t);
    v2f b1 = *(const v2f*)(b1row + 4 * t);
    acc0 = __builtin_amdgcn_wmma_f32_16x16x4_f32(false, a, false, b0, (short)0, acc0,
                                                 false, false);
    acc1 = __builtin_amdgcn_wmma_f32_16x16x4_f32(false, a, false, b1, (short)0, acc1,
                                                 false, false);
  }
  int n = lane & 15;
  int mhi = (lane >> 4) << 3;
  #pragma unroll
  for (int v = 0; v < 8; ++v) {
    int m = (wave << 4) + mhi + v;
    if (m < R) {
      int h = m & 3;                      // tile base is multiple of 16 -> h = (mhi+v)&3
      out[(long)m * 32 + n]      = acc0[v] + bias[h * 32 + n];
      out[(long)m * 32 + 16 + n] = acc1[v] + bias[h * 32 + 16 + n];
    }
  }
}

// ---------------------------------------------------------------------------
extern "C" void kernel_launch(void* const* d_in, const int* in_sizes, int n_in,
                              void* d_out, int out_size, void* d_ws, size_t ws_size,
                              hipStream_t stream) {
  const float* feat   = (const float*)d_in[0];  // [N,100]
  const float* nf     = (const float*)d_in[1];  // [N,64]
  const int*   src    = (const int*)d_in[2];    // [E]
  const int*   dst    = (const int*)d_in[3];    // [E]
  const float* W_type = (const float*)d_in[4];  // [64,100]
  const float* W_fc   = (const float*)d_in[5];  // [128,64]
  const float* attn_l = (const float*)d_in[6];  // [128]
  const float* attn_r = (const float*)d_in[7];  // [128]
  const float* W_out  = (const float*)d_in[8];  // [32,64]
  const float* bias   = (const float*)d_in[9];  // [128]
  float* out = (float*)d_out;                   // [N,4,32]

  const int N = in_sizes[0] / 100;
  const int E = in_sizes[2];
  (void)n_in; (void)out_size; (void)ws_size;

  // workspace layout (floats), 256B-aligned regions
  float* ws = (float*)d_ws;
  size_t o = 0;
  auto take = [&](size_t nelem) { size_t r = o; o += (nelem + 63) & ~(size_t)63; return r; };
  float*    fragB = ws + take(1600);
  float*    el    = ws + take((size_t)N * 4);
  float*    er    = ws + take((size_t)N * 4);
  unsigned* maxu  = (unsigned*)(ws + take((size_t)N * 4));
  float*    denom = ws + take((size_t)N * 4);
  float*    exb   = ws + take((size_t)E * 4);
  float*    agg   = ws + take((size_t)N * 256);

  precompute_kernel<<<1, 256, 0, stream>>>(W_type, W_fc, attn_l, attn_r, fragB);

  init_kernel<<<4096, 256, 0, stream>>>(agg, denom, maxu, (long)N * 256, N * 4);

  {
    int ntiles = (N + 15) >> 4;
    int blocks = (ntiles * 32 + 255) / 256;
    node_elr_kernel<<<blocks, 256, 0, stream>>>(feat, fragB, el, er, N);
  }

  edge_logits_kernel<<<(E + 255) / 256, 256, 0, stream>>>(src, dst, el, er, exb, maxu, E);
  edge_exp_kernel<<<(E + 255) / 256, 256, 0, stream>>>(dst, exb, maxu, denom, E);

  {
    long total = (long)E * 64;
    long blocks = (total + 255) / 256;
    edge_agg_kernel<<<(unsigned)blocks, 256, 0, stream>>>(src, dst, exb, denom, nf, agg, total);
  }

  {
    int R = N * 4;
    int ntiles = (R + 15) >> 4;
    int blocks = (ntiles * 32 + 255) / 256;
    out_gemm_kernel<<<blocks, 256, 0, stream>>>(agg, W_out, bias, out, R);
  }
}